// GATLayer_7000796693165
// MI455X (gfx1250) — compile-verified
//
#include <hip/hip_runtime.h>
#include <hip/hip_bf16.h>

typedef __attribute__((ext_vector_type(2))) float v2f;
typedef __attribute__((ext_vector_type(8))) float v8f;

#define IN_DIM 128
#define HEADS 8
#define HEAD_DIM 16
#define NEG_SLOPE 0.2f

__device__ __forceinline__ float leaky(float v) {
    return v > 0.f ? v : NEG_SLOPE * v;
}

// Order-preserving float -> uint mapping (monotonic), for atomicMax on floats.
__device__ __forceinline__ unsigned f2key(float f) {
    unsigned b = __float_as_uint(f);
    return (b & 0x80000000u) ? ~b : (b | 0x80000000u);
}
__device__ __forceinline__ float key2f(unsigned k) {
    unsigned b = (k & 0x80000000u) ? (k ^ 0x80000000u) : ~k;
    return __uint_as_float(b);
}

// ---------------------------------------------------------------------------
// Init: emax keys to 0 (== "below any finite float"), denom and out to 0.
// ---------------------------------------------------------------------------
__global__ void gat_init(unsigned* __restrict__ emax, float* __restrict__ denom,
                         float* __restrict__ out, int nN) {
    int t = blockIdx.x * blockDim.x + threadIdx.x;
    if (t < nN * HEADS) { emax[t] = 0u; denom[t] = 0.f; }
    if (t < nN * IN_DIM) out[t] = 0.f;
}

// ---------------------------------------------------------------------------
// Fused projection + attention coefficients.
// Block = 256 threads (8 waves). Block b owns rows [16b, 16b+16).
// Wave w owns output columns [16w, 16w+16) == head w.
// WMMA: D(16x16,f32) += A(16x4,f32) x B(4x16,f32), 32 K-steps over IN_DIM=128.
//
// fp32 A 16x4 layout: lanes 0-15 -> {K=k0,k0+1} in VGPR0/1, lanes 16-31 -> {k0+2,k0+3}.
// fp32 C/D 16x16 layout: VGPR r: lanes 0-15 = (M=r, N=lane), lanes 16-31 = (M=8+r).
// ---------------------------------------------------------------------------
__global__ void gat_proj_wmma(const float* __restrict__ x, const float* __restrict__ W,
                              const float* __restrict__ a_src, const float* __restrict__ a_dst,
                              float* __restrict__ h, float* __restrict__ as_out,
                              float* __restrict__ ad_out) {
    const int lane = threadIdx.x & 31;
    const int head = threadIdx.x >> 5;   // 0..7 == N-tile == head
    const int m0   = blockIdx.x * 16;
    const int half = lane >> 4;          // 0: K pair {0,1}; 1: K pair {2,3}
    const int l15  = lane & 15;
    const int arow = m0 + l15;
    const int col  = head * HEAD_DIM + l15;

    const float2* __restrict__ xrow = (const float2*)(x + (size_t)arow * IN_DIM);
    const float*  __restrict__ wcol = W + (size_t)(half * 2) * IN_DIM + col;

    v8f c = {};
#pragma unroll 4
    for (int k0 = 0; k0 < IN_DIM; k0 += 4) {
        float2 av = xrow[(k0 >> 1) + half];     // x[arow][k0 + 2*half + {0,1}]
        v2f a, b;
        a.x = av.x;
        a.y = av.y;
        b.x = wcol[(size_t)k0 * IN_DIM];        // W[k0 + 2*half    ][col]
        b.y = wcol[(size_t)(k0 + 1) * IN_DIM];  // W[k0 + 2*half + 1][col]
        c = __builtin_amdgcn_wmma_f32_16x16x4_f32(
                /*neg_a=*/false, a, /*neg_b=*/false, b,
                /*c_mod=*/(short)0, c, /*reuse_a=*/false, /*reuse_b=*/false);
    }

    // Per-head attention vectors for this wave's 16 columns (d = l15).
    const float asv = a_src[head * HEAD_DIM + l15];
    const float adv = a_dst[head * HEAD_DIM + l15];

#pragma unroll
    for (int r = 0; r < 8; ++r) {
        const int row = m0 + r + half * 8;
        const float v = c[r];
        h[(size_t)row * IN_DIM + col] = v;
        float s = v * asv;
        float d = v * adv;
        // Reduce across the 16 lanes that share this matrix row (wave32).
#pragma unroll
        for (int off = 8; off > 0; off >>= 1) {
            s += __shfl_xor(s, off, 32);
            d += __shfl_xor(d, off, 32);
        }
        if (l15 == 0) {
            as_out[row * HEADS + head] = s;
            ad_out[row * HEADS + head] = d;
        }
    }
}

// ---------------------------------------------------------------------------
// Pass 1: per-(edge, head) logit, atomic segment-max over dst.
// Edges e >= nE are the self-loops (src = dst = e - nE).
// ---------------------------------------------------------------------------
__global__ void gat_edge_max(const int* __restrict__ ei, int nE, int nN,
                             const float* __restrict__ as, const float* __restrict__ ad,
                             unsigned* __restrict__ emax) {
    int t = blockIdx.x * blockDim.x + threadIdx.x;
    int total = (nE + nN) * HEADS;
    if (t >= total) return;
    int e = t >> 3, head = t & 7;
    int s, d;
    if (e < nE) { s = ei[e]; d = ei[nE + e]; } else { s = d = e - nE; }
    float v = leaky(as[s * HEADS + head] + ad[d * HEADS + head]);
    atomicMax(&emax[d * HEADS + head], f2key(v));
}

// ---------------------------------------------------------------------------
// Pass 2: denom[dst,h] += exp(logit - max[dst,h]).
// ---------------------------------------------------------------------------
__global__ void gat_edge_denom(const int* __restrict__ ei, int nE, int nN,
                               const float* __restrict__ as, const float* __restrict__ ad,
                               const unsigned* __restrict__ emax, float* __restrict__ denom) {
    int t = blockIdx.x * blockDim.x + threadIdx.x;
    int total = (nE + nN) * HEADS;
    if (t >= total) return;
    int e = t >> 3, head = t & 7;
    int s, d;
    if (e < nE) { s = ei[e]; d = ei[nE + e]; } else { s = d = e - nE; }
    float v = leaky(as[s * HEADS + head] + ad[d * HEADS + head]);
    float m = key2f(emax[d * HEADS + head]);
    atomicAdd(&denom[d * HEADS + head], __expf(v - m));
}

// ---------------------------------------------------------------------------
// Pass 3: out[dst] += alpha * h[src]. One wave per edge; lane l handles
// channels [4l, 4l+4) (all inside head l/4): float4 gather + 4 float atomics.
// ---------------------------------------------------------------------------
__global__ void gat_scatter(const int* __restrict__ ei, int nE, int nN,
                            const float* __restrict__ as, const float* __restrict__ ad,
                            const unsigned* __restrict__ emax, const float* __restrict__ denom,
                            const float* __restrict__ h, float* __restrict__ out) {
    int wid  = (blockIdx.x * blockDim.x + threadIdx.x) >> 5;
    int lane = threadIdx.x & 31;
    if (wid >= nE + nN) return;
    int s, d;
    if (wid < nE) { s = ei[wid]; d = ei[nE + wid]; } else { s = d = wid - nE; }
    int head = lane >> 2;
    int c0   = lane * 4;
    float v = leaky(as[s * HEADS + head] + ad[d * HEADS + head]);
    float m = key2f(emax[d * HEADS + head]);
    float alpha = __expf(v - m) / (denom[d * HEADS + head] + 1e-16f);
    float4 hv = *(const float4*)(h + (size_t)s * IN_DIM + c0);
    float* o = out + (size_t)d * IN_DIM + c0;
    atomicAdd(o + 0, alpha * hv.x);
    atomicAdd(o + 1, alpha * hv.y);
    atomicAdd(o + 2, alpha * hv.z);
    atomicAdd(o + 3, alpha * hv.w);
}

// ---------------------------------------------------------------------------
// Final: out = relu(out + bias).
// ---------------------------------------------------------------------------
__global__ void gat_finish(float* __restrict__ out, const float* __restrict__ bias, int n) {
    int t = blockIdx.x * blockDim.x + threadIdx.x;
    if (t >= n) return;
    float v = out[t] + bias[t & (IN_DIM - 1)];
    out[t] = v > 0.f ? v : 0.f;
}

extern "C" void kernel_launch(void* const* d_in, const int* in_sizes, int n_in,
                              void* d_out, int out_size, void* d_ws, size_t ws_size,
                              hipStream_t stream) {
    const float* x     = (const float*)d_in[0];
    const int*   ei    = (const int*)d_in[1];
    const float* W     = (const float*)d_in[2];
    const float* a_src = (const float*)d_in[3];
    const float* a_dst = (const float*)d_in[4];
    const float* bias  = (const float*)d_in[5];
    float* out = (float*)d_out;

    const int nN = in_sizes[0] / IN_DIM; // 100000
    const int nE = in_sizes[1] / 2;      // 1600000

    // Workspace layout (floats): h[N*128] | as[N*8] | ad[N*8] | denom[N*8] | emax[N*8]
    float*    h   = (float*)d_ws;
    float*    as_ = h + (size_t)nN * IN_DIM;
    float*    ad_ = as_ + (size_t)nN * HEADS;
    float*    den = ad_ + (size_t)nN * HEADS;
    unsigned* em  = (unsigned*)(den + (size_t)nN * HEADS);

    const int totOut = nN * IN_DIM;
    gat_init<<<(totOut + 255) / 256, 256, 0, stream>>>(em, den, out, nN);

    gat_proj_wmma<<<nN / 16, 256, 0, stream>>>(x, W, a_src, a_dst, h, as_, ad_);

    const int totEH = (nE + nN) * HEADS;
    gat_edge_max<<<(totEH + 255) / 256, 256, 0, stream>>>(ei, nE, nN, as_, ad_, em);
    gat_edge_denom<<<(totEH + 255) / 256, 256, 0, stream>>>(ei, nE, nN, as_, ad_, em, den);

    const int nWaves = nE + nN; // one wave per edge
    gat_scatter<<<(nWaves + 7) / 8, 256, 0, stream>>>(ei, nE, nN, as_, ad_, em, den, h, out);

    gat_finish<<<(totOut + 255) / 256, 256, 0, stream>>>(out, bias, totOut);
}